// Loss_38259568673419
// MI455X (gfx1250) — compile-verified
//
#include <hip/hip_runtime.h>
#include <hip/hip_bf16.h>

// ---------------------------------------------------------------------------
// SSD-style loss on MI455X (gfx1250, wave32).
//
// Roofline: ~22MB traffic -> ~1us @ 23.3TB/s (inputs stay resident in the
// 192MB L2 across graph replays, so default RT caching is optimal). The naive
// pair loop is VALU-bound (~6x over memory), so the hot loop is stripped to a
// division-free IoU threshold test building a 24-bit positive bitmask, with
// the rare smooth-L1 work branched out, and the g-loop bounded by
// num_objects[b] (uniform per block).
//
//   Stage 1: grid (64, 16) x 256 threads; gt[b] staged in LDS; 4 anchors per
//            thread held in registers, g-loop outer. Per-wave reduction via
//            V_WMMA_F32_16X16X4_F32 (exact f32) + xor shuffles, cross-wave
//            via LDS in fixed order -> ws partials.
//   Stage 2: 1 wave; lane b<16 sums its 64 partials serially, applies
//            per-image normalization, xor tree -> (total, class, coord).
// No atomics -> bit-identical across graph replays.
// ---------------------------------------------------------------------------

typedef __attribute__((ext_vector_type(2))) float v2f;
typedef __attribute__((ext_vector_type(8))) float v8f;

#define NG 24
#define NA 65536
#define NB 16
#define CHUNKS 64               // blocks per image
#define APB 1024                // anchors per block
#define THREADS 256             // 8 waves
#define APT 4                   // anchors per thread
#define KIOU (0.55f / 1.55f)    // iou>=0.55  <=>  inter >= KIOU*(areaA+areaG)

__device__ __forceinline__ float smooth_l1(float d) {
    float ad = fabsf(d);
    return (ad < 1.0f) ? 0.5f * d * d : ad - 0.5f;
}

// Exact f32 cross-lane (wave32) sum via V_WMMA_F32_16X16X4_F32:
//   A = ones(16x4), B vgpr0 = per-lane value, B vgpr1 = 0.
//   B rows are lane-striped within a VGPR, so D[m,n] = v[n] + v[n+16] for all
//   m; D vgpr0 at lane L holds colsum(L%16); 4 xor-shuffles finish 16->1.
// EXEC must be all ones at the call site (uniform control flow).
__device__ __forceinline__ float wave_sum_wmma(float v) {
    v2f a; a.x = 1.0f; a.y = 1.0f;
    v2f b; b.x = v;    b.y = 0.0f;
    v8f c = {};
    c = __builtin_amdgcn_wmma_f32_16x16x4_f32(
            false, a, false, b, (short)0, c, false, false);
    float s = c[0];                    // v[lane%16] + v[lane%16 + 16]
    s += __shfl_xor(s, 8, 32);
    s += __shfl_xor(s, 4, 32);
    s += __shfl_xor(s, 2, 32);
    s += __shfl_xor(s, 1, 32);
    return s;                          // full wave sum in every lane
}

__global__ void __launch_bounds__(THREADS)
ssd_loss_partials(const float* __restrict__ boxes,    // [B, A, 4]
                  const float* __restrict__ classes,  // [B, A]
                  const float* __restrict__ anchors,  // [A, 4]
                  const float* __restrict__ gt,       // [B, G, 4]
                  const int*   __restrict__ nobj,     // [B]
                  float*       __restrict__ ws) {     // [B, CHUNKS, 3]
    __shared__ float4 s_gt[NG];
    __shared__ float  s_thr[NG];       // KIOU * area_g
    __shared__ int    s_n;
    __shared__ float  s_wave[8][3];

    const int b     = blockIdx.y;
    const int chunk = blockIdx.x;
    const int tid   = threadIdx.x;

    if (tid < NG) {
        float4 g = ((const float4*)gt)[b * NG + tid];
        s_gt[tid]  = g;
        s_thr[tid] = KIOU * ((g.z - g.x) * (g.w - g.y));
    }
    if (tid == 0) s_n = nobj[b];
    __syncthreads();

    const int n_valid = s_n;           // uniform per block, 1..24
    const int a_base  = chunk * APB;

    // Load APT anchors' data into registers.
    float4 an[APT], bx[APT];
    float  x[APT], cA[APT];
    #pragma unroll
    for (int i = 0; i < APT; ++i) {
        const int a = a_base + i * THREADS + tid;
        an[i] = ((const float4*)anchors)[a];
        bx[i] = ((const float4*)boxes)[(size_t)b * NA + a];
        x[i]  = classes[(size_t)b * NA + a];
        cA[i] = KIOU * ((an[i].z - an[i].x) * (an[i].w - an[i].y));
    }

    // --- Hot loop: division-free IoU threshold -> positive bitmasks -------
    unsigned mask[APT] = {0u, 0u, 0u, 0u};
    for (int g = 0; g < n_valid; ++g) {
        const float4 gb  = s_gt[g];
        const float  thr = s_thr[g];
        #pragma unroll
        for (int i = 0; i < APT; ++i) {
            float iw = fminf(an[i].z, gb.z) - fmaxf(an[i].x, gb.x);
            float ih = fminf(an[i].w, gb.w) - fmaxf(an[i].y, gb.y);
            iw = fmaxf(iw, 0.0f);
            ih = fmaxf(ih, 0.0f);
            const bool p = (iw * ih) >= (cA[i] + thr);
            mask[i] |= (p ? 1u : 0u) << g;
        }
    }

    // --- Rare path: smooth-L1 only for positive pairs ---------------------
    float acc_cls = 0.0f, acc_crd = 0.0f, acc_cnt = 0.0f;
    #pragma unroll
    for (int i = 0; i < APT; ++i) {
        const float t = mask[i] ? 1.0f : 0.0f;
        acc_cls += fmaxf(x[i], 0.0f) - x[i] * t +
                   log1pf(expf(-fabsf(x[i])));
        if (mask[i]) {                 // rare; execz-skipped for most waves
            acc_cnt += (float)__popc(mask[i]);
            unsigned m = mask[i];
            while (m) {
                const int g = __builtin_ctz(m);
                m &= m - 1u;
                const float4 gb = s_gt[g];
                acc_crd += smooth_l1(bx[i].x - gb.x) +
                           smooth_l1(bx[i].y - gb.y) +
                           smooth_l1(bx[i].z - gb.z) +
                           smooth_l1(bx[i].w - gb.w);
            }
        }
    }

    // --- wave32 reduction via WMMA (uniform control flow, EXEC all ones) --
    const float wcls = wave_sum_wmma(acc_cls);
    const float wcrd = wave_sum_wmma(acc_crd);
    const float wcnt = wave_sum_wmma(acc_cnt);

    const int wave = tid >> 5;
    const int lane = tid & 31;
    if (lane == 0) {
        s_wave[wave][0] = wcls;
        s_wave[wave][1] = wcrd;
        s_wave[wave][2] = wcnt;
    }
    __syncthreads();

    if (tid == 0) {
        float tc = 0.0f, tr = 0.0f, tn = 0.0f;
        #pragma unroll
        for (int w = 0; w < 8; ++w) {  // fixed order -> deterministic
            tc += s_wave[w][0];
            tr += s_wave[w][1];
            tn += s_wave[w][2];
        }
        float* p = ws + ((size_t)b * CHUNKS + chunk) * 3;
        p[0] = tc; p[1] = tr; p[2] = tn;
    }
}

__global__ void __launch_bounds__(32)
ssd_loss_finalize(const float* __restrict__ ws,    // [B, CHUNKS, 3]
                  float*       __restrict__ out) { // [3]: total, class, coord
    const int tid = threadIdx.x;
    float cc = 0.0f, rc = 0.0f;
    if (tid < NB) {
        float cls = 0.0f, crd = 0.0f, cnt = 0.0f;
        for (int k = 0; k < CHUNKS; ++k) {          // fixed order
            const float* p = ws + ((size_t)tid * CHUNKS + k) * 3;
            cls += p[0]; crd += p[1]; cnt += p[2];
        }
        const float num_pos = (cnt > 0.0f) ? cnt : 1.0f;
        cc = cls / num_pos;
        rc = (cnt > 0.0f) ? crd / (4.0f * cnt) : 0.0f;
    }
    #pragma unroll
    for (int m = 16; m >= 1; m >>= 1) {            // lanes >=16 contribute 0
        cc += __shfl_xor(cc, m, 32);
        rc += __shfl_xor(rc, m, 32);
    }
    if (tid == 0) {
        const float class_loss = cc / (float)NB;
        const float coord_loss = rc / (float)NB;
        out[0] = class_loss + coord_loss;
        out[1] = class_loss;
        out[2] = coord_loss;
    }
}

extern "C" void kernel_launch(void* const* d_in, const int* in_sizes, int n_in,
                              void* d_out, int out_size, void* d_ws, size_t ws_size,
                              hipStream_t stream) {
    // setup_inputs() order:
    //   0: batch_boxes      [B, A, 4] f32
    //   1: batch_classes    [B, A]    f32
    //   2: anchors          [A, 4]    f32
    //   3: batch_gt         [B, G, 4] f32
    //   4: batch_num_objects[B]       i32
    const float* boxes   = (const float*)d_in[0];
    const float* classes = (const float*)d_in[1];
    const float* anchors = (const float*)d_in[2];
    const float* gt      = (const float*)d_in[3];
    const int*   nobj    = (const int*)d_in[4];
    float*       out     = (float*)d_out;
    float*       ws      = (float*)d_ws;   // needs 16*64*3*4 = 12 KB

    dim3 grid(CHUNKS, NB);
    ssd_loss_partials<<<grid, THREADS, 0, stream>>>(boxes, classes, anchors,
                                                    gt, nobj, ws);
    ssd_loss_finalize<<<1, 32, 0, stream>>>(ws, out);
}